// DefAgg_71786083385912
// MI455X (gfx1250) — compile-verified
//
#include <hip/hip_runtime.h>
#include <hip/hip_bf16.h>

// Deformable aggregation for MI455X (gfx1250).
// Memory/gather-bound op (~44MB traffic -> ~2us at 23.3 TB/s; only ~0.3 GFLOP),
// so the win is data movement: stage each (batch, group) 56x56x8 input plane
// (98 KB) into the 320KB WGP LDS in channel-minor layout via the CDNA5 async
// global->LDS DMA path (per-lane LDS destinations give a free NCHW->NHWC tile
// transpose), then bilinear-gather from LDS with 128-bit ds loads
// (9 taps x 4 corners x 2 ds_load_b128 per pixel).
// WMMA is deliberately NOT used: the per-pixel modulation weights make the
// K=9 aggregation a per-pixel 8x9 matvec (block-diagonal if forced into a
// GEMM, ~16x wasted lanes) - VALU FMA is the right tool on this op.

#define BN   16
#define CN   64
#define HN   56
#define WN   56
#define GR   8
#define CPG  8            // channels per group
#define KK   9
#define PADP 1
#define HWN  (HN * WN)    // 3136
#define SPLIT 2
#define PIX_PER_WG (HWN / SPLIT)   // 1568
#define NTHREADS 256

typedef __attribute__((address_space(1))) int glb_i32;
typedef __attribute__((address_space(3))) int lds_i32;

__global__ __launch_bounds__(NTHREADS)
void defagg_kernel(const float* __restrict__ in,
                   const float* __restrict__ off,
                   const float* __restrict__ wgt,
                   float* __restrict__ out)
{
    // [pixel][channel] channel-minor plane of one group: 56*56*8 floats = 98KB
    __shared__ float smem[HWN * CPG];

    const int wg   = blockIdx.x;
    const int half = wg % SPLIT;
    const int g    = (wg / SPLIT) % GR;
    const int b    = wg / (SPLIT * GR);
    const int tid  = threadIdx.x;

    const float* src = in + ((size_t)(b * CN + g * CPG)) * HWN;

    // ---- Stage input plane into LDS, transposed to channel-minor ----------
    // Channel-outer / pixel-inner: coalesced global reads, no divisions in
    // the DMA-issue loop; each lane writes its own scattered LDS address.
#if __has_builtin(__builtin_amdgcn_global_load_async_to_lds_b32)
    {
#pragma unroll
        for (int c = 0; c < CPG; ++c) {
            const float* srcc = src + c * HWN;
            float*       dstc = &smem[c];
            for (int p = tid; p < HWN; p += NTHREADS) {
                __builtin_amdgcn_global_load_async_to_lds_b32(
                    (glb_i32*)(srcc + p),
                    (lds_i32*)(dstc + p * CPG),
                    0, 0);
            }
        }
        asm volatile("s_wait_asynccnt 0" ::: "memory");
    }
#else
    for (int c = 0; c < CPG; ++c) {
        const float* srcc = src + c * HWN;
        for (int p = tid; p < HWN; p += NTHREADS)
            smem[p * CPG + c] = srcc[p];
    }
#endif
    __syncthreads();

    // ---- Compute ----------------------------------------------------------
    const float* offb = off + (size_t)b * (2 * KK) * HWN;
    const float* wgtb = wgt + ((size_t)(b * GR + g)) * KK * HWN;
    float*       outb = out + ((size_t)(b * CN + g * CPG)) * HWN;

    const int pend = (half + 1) * PIX_PER_WG;
    for (int p = half * PIX_PER_WG + tid; p < pend; p += NTHREADS) {
        const int y = p / WN;
        const int x = p - y * WN;

        float4 acc0 = make_float4(0.f, 0.f, 0.f, 0.f);
        float4 acc1 = make_float4(0.f, 0.f, 0.f, 0.f);

#pragma unroll
        for (int k = 0; k < KK; ++k) {
            const int ky = k / 3;
            const int kx = k - ky * 3;
            const float oy = offb[(2 * k + 0) * HWN + p];
            const float ox = offb[(2 * k + 1) * HWN + p];
            const float wk = wgtb[k * HWN + p];

            const float py = (float)(y - PADP + ky) + oy;
            const float px = (float)(x - PADP + kx) + ox;
            const float y0f = floorf(py);
            const float x0f = floorf(px);
            const float wy1 = py - y0f, wx1 = px - x0f;
            const float wy0 = 1.f - wy1, wx0 = 1.f - wx1;
            const int y0 = (int)y0f;
            const int x0 = (int)x0f;

#pragma unroll
            for (int dy = 0; dy < 2; ++dy) {
                const int   yc = y0 + dy;
                const float wy = dy ? wy1 : wy0;
                const bool  vy = (yc >= 0) && (yc < HN);
                const int   yi = min(max(yc, 0), HN - 1);
#pragma unroll
                for (int dx = 0; dx < 2; ++dx) {
                    const int   xc = x0 + dx;
                    const float wx = dx ? wx1 : wx0;
                    const bool  v  = vy && (xc >= 0) && (xc < WN);
                    const int   xi = min(max(xc, 0), WN - 1);
                    const float cw = v ? (wy * wx * wk) : 0.f;

                    const float4* lp = (const float4*)&smem[(yi * WN + xi) * CPG];
                    const float4 a = lp[0];
                    const float4 c = lp[1];
                    acc0.x = fmaf(cw, a.x, acc0.x);
                    acc0.y = fmaf(cw, a.y, acc0.y);
                    acc0.z = fmaf(cw, a.z, acc0.z);
                    acc0.w = fmaf(cw, a.w, acc0.w);
                    acc1.x = fmaf(cw, c.x, acc1.x);
                    acc1.y = fmaf(cw, c.y, acc1.y);
                    acc1.z = fmaf(cw, c.z, acc1.z);
                    acc1.w = fmaf(cw, c.w, acc1.w);
                }
            }
        }

        float* op = outb + p;
        op[0 * HWN] = acc0.x;
        op[1 * HWN] = acc0.y;
        op[2 * HWN] = acc0.z;
        op[3 * HWN] = acc0.w;
        op[4 * HWN] = acc1.x;
        op[5 * HWN] = acc1.y;
        op[6 * HWN] = acc1.z;
        op[7 * HWN] = acc1.w;
    }
}

extern "C" void kernel_launch(void* const* d_in, const int* in_sizes, int n_in,
                              void* d_out, int out_size, void* d_ws, size_t ws_size,
                              hipStream_t stream)
{
    (void)in_sizes; (void)n_in; (void)out_size; (void)d_ws; (void)ws_size;
    const float* in  = (const float*)d_in[0];
    const float* off = (const float*)d_in[1];
    const float* wgt = (const float*)d_in[2];
    float* out = (float*)d_out;

    dim3 grid(BN * GR * SPLIT);   // 256 workgroups, one (b, g, half-plane) each
    dim3 block(NTHREADS);         // 8 wave32s
    defagg_kernel<<<grid, block, 0, stream>>>(in, off, wgt, out);
}